// LinearCRF_32590211842359
// MI455X (gfx1250) — compile-verified
//
#include <hip/hip_runtime.h>

#define HID  512
#define TAGS 48
#define TT   512
#define BB   256

typedef __attribute__((ext_vector_type(2))) float v2f;
typedef __attribute__((ext_vector_type(8))) float v8f;

// ---------------------------------------------------------------------------
// Kernel 1: emission GEMM via V_WMMA_F32_16X16X4_F32.
// E[m, k] = sum_h hiddens[m, h] * W[k, h] + bias[k],  m = t*B + b (rows are
// contiguous with stride HID, so we treat hiddens as a [T*B, HID] matrix).
// One wave computes a 16-row x 48-tag tile using 3 f32 accumulators.
//
// f32 16x16x4 operand layout (ISA 7.12.2):
//   A (16x4):  lane L<16:  VGPR0=A[M=L][K=0], VGPR1=A[M=L][K=1]
//              lane L>=16: VGPR0=A[M=L-16][K=2], VGPR1=A[M=L-16][K=3]
//   B (4x16):  lane L<16:  VGPR0=B[K=0][N=L], VGPR1=B[K=1][N=L]
//              lane L>=16: VGPR0=B[K=2][N=L-16], VGPR1=B[K=3][N=L-16]
//   C/D (16x16): VGPR v: lanes 0-15 -> M=v, N=lane; lanes 16-31 -> M=v+8.
// ---------------------------------------------------------------------------
__global__ __launch_bounds__(256) void linearcrf_emissions_wmma(
    const float* __restrict__ hid,   // [T*B, HID]
    const float* __restrict__ W,     // [TAGS, HID]
    const float* __restrict__ bias,  // [TAGS]
    float* __restrict__ em)          // [T*B, TAGS]
{
    const int lane      = threadIdx.x & 31;
    const int waveInBlk = threadIdx.x >> 5;
    const long tileRow  = (long)blockIdx.x * 8 + waveInBlk;  // one 16-row tile per wave
    const long m0       = tileRow * 16;

    const int lrow  = lane & 15;          // M index (A) / N index (B,C)
    const int khalf = (lane >> 4) << 1;   // 0 for lanes 0-15, 2 for lanes 16-31

    const float* arow  = hid + (m0 + lrow) * (long)HID + khalf;
    const float* b0row = W + (0 * 16 + lrow) * (long)HID + khalf;
    const float* b1row = W + (1 * 16 + lrow) * (long)HID + khalf;
    const float* b2row = W + (2 * 16 + lrow) * (long)HID + khalf;

    v8f c0 = {}, c1 = {}, c2 = {};
    #pragma unroll 4
    for (int h = 0; h < HID; h += 4) {
        v2f a  = *(const v2f*)(arow + h);
        v2f b0 = *(const v2f*)(b0row + h);
        v2f b1 = *(const v2f*)(b1row + h);
        v2f b2 = *(const v2f*)(b2row + h);
        c0 = __builtin_amdgcn_wmma_f32_16x16x4_f32(false, a, false, b0, (short)0, c0, false, false);
        c1 = __builtin_amdgcn_wmma_f32_16x16x4_f32(false, a, false, b1, (short)0, c1, false, false);
        c2 = __builtin_amdgcn_wmma_f32_16x16x4_f32(false, a, false, b2, (short)0, c2, false, false);
    }

    const int mhalf = (lane >> 4) << 3;  // 0 or 8
    const float bs0 = bias[0 * 16 + lrow];
    const float bs1 = bias[1 * 16 + lrow];
    const float bs2 = bias[2 * 16 + lrow];
    #pragma unroll
    for (int v = 0; v < 8; ++v) {
        const long r = (m0 + mhalf + v) * (long)TAGS;
        em[r + 0 * 16 + lrow] = c0[v] + bs0;
        em[r + 1 * 16 + lrow] = c1[v] + bs1;
        em[r + 2 * 16 + lrow] = c2[v] + bs2;
    }
}

// ---------------------------------------------------------------------------
// Kernel 2: masked Viterbi recursion + in-LDS backtrace.
// One 64-thread workgroup per sentence b; thread k owns curr-tag k and keeps
// transition[k, :] (48 floats) in VGPRs. fv is double-buffered in LDS (one
// barrier per step). Backpointers are byte-packed into LDS (511*48 = 24 KB,
// well under CDNA5's 320 KB/WGP) so the sequential backtrace is a chain of
// DS loads instead of 511 dependent global-memory round trips.
// ---------------------------------------------------------------------------
__global__ __launch_bounds__(64) void linearcrf_viterbi(
    const float* __restrict__ em,          // [T*B, TAGS]
    const long long* __restrict__ lens,    // [B] (int64)
    const float* __restrict__ trans,       // [TAGS, TAGS]  trans[curr, prev]
    float* __restrict__ out)               // [B] scores, then [T, B] paths (as f32)
{
    __shared__ float fvbuf[2][TAGS];
    __shared__ unsigned char bpL[(TT - 1) * TAGS];

    const int b   = blockIdx.x;
    const int k   = threadIdx.x;
    const int len = (int)lens[b];

    float tr[TAGS];
    if (k < TAGS) {
        #pragma unroll
        for (int p = 0; p < TAGS; ++p) tr[p] = trans[k * TAGS + p];
        fvbuf[0][k] = em[(long)b * TAGS + k];   // fv0 = emissions[t=0]
    }
    __syncthreads();

    int cur = 0;
    #pragma unroll 1
    for (int t = 1; t < TT; ++t) {
        if (k < TAGS) {
            const float fvk = fvbuf[cur][k];
            float best = -3.402823466e38f;
            int   bi   = 0;
            #pragma unroll
            for (int p = 0; p < TAGS; ++p) {
                const float s = fvbuf[cur][p] + tr[p];
                if (s > best) { best = s; bi = p; }   // first-max (argmax semantics)
            }
            const bool  valid = (t < len);
            const float e     = em[((long)t * BB + b) * (long)TAGS + k];
            fvbuf[cur ^ 1][k]     = valid ? (best + e) : fvk;
            bpL[(t - 1) * TAGS + k] = valid ? (unsigned char)bi : (unsigned char)k;
        }
        __syncthreads();
        cur ^= 1;
    }

    if (k == 0) {
        float bs = -3.402823466e38f;
        int   bl = 0;
        for (int p = 0; p < TAGS; ++p) {
            const float v = fvbuf[cur][p];
            if (v > bs) { bs = v; bl = p; }
        }
        out[b] = bs;                                      // best_score
        out[BB + (long)(TT - 1) * BB + b] = (float)bl;    // paths[T-1]
        int c = bl;
        for (int i = TT - 2; i >= 0; --i) {               // backtrace in LDS
            c = (int)bpL[i * TAGS + c];
            out[BB + (long)i * BB + b] = (float)c;
        }
    }
}

extern "C" void kernel_launch(void* const* d_in, const int* in_sizes, int n_in,
                              void* d_out, int out_size, void* d_ws, size_t ws_size,
                              hipStream_t stream) {
    const float*     hiddens = (const float*)d_in[0];     // [T, B, HID]
    const long long* lens    = (const long long*)d_in[1]; // [B]
    const float*     W       = (const float*)d_in[2];     // [TAGS, HID]
    const float*     bias    = (const float*)d_in[3];     // [TAGS]
    const float*     trans   = (const float*)d_in[4];     // [TAGS, TAGS]
    float*           out     = (float*)d_out;             // [B] + [T, B]
    float*           em      = (float*)d_ws;              // [T*B, TAGS] = 25.2 MB

    (void)in_sizes; (void)n_in; (void)out_size; (void)ws_size;

    // GEMM: T*B/16 = 8192 wave-tiles, 8 waves per 256-thread block -> 1024 blocks.
    linearcrf_emissions_wmma<<<dim3(1024), dim3(256), 0, stream>>>(hiddens, W, bias, em);

    // Viterbi: one workgroup per sentence.
    linearcrf_viterbi<<<dim3(BB), dim3(64), 0, stream>>>(em, lens, trans, out);
}